// MaskedScaleDotProductModule_31765578121670
// MI455X (gfx1250) — compile-verified
//
#include <hip/hip_runtime.h>
#include <math.h>

#define S_LEN   2048
#define D_HEAD  64
#define N_BH    64            // B*H = 4*16
#define SCALE   0.125f        // 1/sqrt(64)
#define GAMMA_F 1.0f
#define NEG_INF_F (-1e9f)

typedef float v2f __attribute__((ext_vector_type(2)));
typedef float v8f __attribute__((ext_vector_type(8)));

// LDS layout (floats): scores[16*2048] | outbuf[16*64] | red[256] | rowmax[16] | rowsum[16]
#define SMEM_FLOATS (16 * S_LEN + 16 * D_HEAD + 256 + 16 + 16)

__global__ __launch_bounds__(256)
void attn_fused_kernel(const float* __restrict__ q,
                       const float* __restrict__ k,
                       const float* __restrict__ v,
                       const int*   __restrict__ mask,
                       float* __restrict__ out,
                       float* __restrict__ attn)
{
    extern __shared__ float smem[];
    float* sc     = smem;                       // 16 x 2048 score / prob slab
    float* outbuf = sc + 16 * S_LEN;            // 16 x 64 AV accumulator
    float* red    = outbuf + 16 * D_HEAD;       // 16 x 16 reduction scratch
    float* rowmax = red + 256;                  // 16
    float* rowsum = rowmax + 16;                // 16

    const int tid  = threadIdx.x;
    const int wave = tid >> 5;                  // 0..7
    const int lane = tid & 31;
    const int half = lane >> 4;                 // 0: lanes 0-15, 1: lanes 16-31
    const int l16  = lane & 15;
    const int kb0  = half * 2;                  // A/B frag K sub-offset

    const int bh   = blockIdx.x >> 7;           // 0..63
    const int tile = blockIdx.x & 127;          // 0..127
    const int b    = bh >> 4;
    const int row0 = tile * 16;

    const long hbase = (long)bh * S_LEN;        // row base into [BH, S, D]
    const float* qh = q + hbase * D_HEAD;
    const float* kh = k + hbase * D_HEAD;
    const float* vh = v + hbase * D_HEAD;
    const int*   mb = mask + (long)b * S_LEN * S_LEN;

    // ---- Load Q tile as 16 chained A-fragments (pre-scaled by 1/8) ----
    // A (16x4 f32): lanes 0-15 -> M=lane, K = 4*kc + {0,1}; lanes 16-31 -> K = 4*kc + {2,3}
    v2f aq[16];
#pragma unroll
    for (int kc = 0; kc < 16; ++kc) {
        const float* p = qh + (long)(row0 + l16) * D_HEAD + 4 * kc + kb0;
        aq[kc].x = p[0] * SCALE;
        aq[kc].y = p[1] * SCALE;
    }

    // ---- Stage 1: scores = (Q/s) @ K^T, masked, into LDS ----
    for (int i = 0; i < 16; ++i) {
        const int key0 = (wave + 8 * i) * 16;
        if (i < 15) {
            // prefetch next K tile for this wave (global_prefetch_b8)
            __builtin_prefetch(kh + (long)(key0 + 128 + l16) * D_HEAD, 0, 0);
        }
        v8f c = {};
#pragma unroll
        for (int kc = 0; kc < 16; ++kc) {
            const float* p = kh + (long)(key0 + l16) * D_HEAD + 4 * kc + kb0;
            v2f bf;
            bf.x = p[0];
            bf.y = p[1];
            c = __builtin_amdgcn_wmma_f32_16x16x4_f32(false, aq[kc], false, bf,
                                                      (short)0, c, false, false);
        }
        // C/D layout: VGPR r -> M = r + 8*half, N = l16
#pragma unroll
        for (int r = 0; r < 8; ++r) {
            const int m   = r + 8 * half;
            const int row = row0 + m;
            const int col = key0 + l16;
            const int mv  = mb[(long)row * S_LEN + col];
            sc[m * S_LEN + col] = mv ? c[r] : NEG_INF_F;
        }
    }
    __syncthreads();

    // ---- Stage 2: row softmax over 2048 cols, write normalized attn ----
    const int srow = tid >> 4;        // 0..15
    const int seg  = tid & 15;        // 128-col segment
    float* rp = sc + srow * S_LEN + seg * 128;

    float mloc = -INFINITY;
    for (int j = 0; j < 128; ++j) mloc = fmaxf(mloc, rp[j]);
    red[srow * 16 + seg] = mloc;
    __syncthreads();
    if (seg == 0) {
        float mm = red[srow * 16];
        for (int j = 1; j < 16; ++j) mm = fmaxf(mm, red[srow * 16 + j]);
        rowmax[srow] = mm;
    }
    __syncthreads();

    const float rm = rowmax[srow];
    float sloc = 0.f;
    for (int j = 0; j < 128; ++j) {
        const float e = __expf(rp[j] - rm);
        rp[j] = e;
        sloc += e;
    }
    red[srow * 16 + seg] = sloc;
    __syncthreads();
    if (seg == 0) {
        float ss = 0.f;
        for (int j = 0; j < 16; ++j) ss += red[srow * 16 + j];
        rowsum[srow] = ss;
    }
    __syncthreads();

    const float inv = GAMMA_F / rowsum[srow];
    float* arow = attn + ((long)(bh * S_LEN + row0 + srow)) * S_LEN + seg * 128;
    for (int j = 0; j < 128; j += 4) {
        float4 pv;
        pv.x = rp[j + 0] * inv;
        pv.y = rp[j + 1] * inv;
        pv.z = rp[j + 2] * inv;
        pv.w = rp[j + 3] * inv;
        rp[j + 0] = pv.x; rp[j + 1] = pv.y; rp[j + 2] = pv.z; rp[j + 3] = pv.w;
        *(float4*)(arow + j) = pv;   // coalesced B128 store of attn
    }
    for (int idx = tid; idx < 16 * D_HEAD; idx += 256) outbuf[idx] = 0.f;
    __syncthreads();

    // ---- Stage 3: AV = P(16x2048) @ V(2048x64) via WMMA, partial per wave ----
    v8f acc[4];
#pragma unroll
    for (int nb = 0; nb < 4; ++nb) { v8f z = {}; acc[nb] = z; }

    for (int i = 0; i < 16; ++i) {
        const int key0 = (wave + 8 * i) * 16;
#pragma unroll
        for (int kc = 0; kc < 4; ++kc) {
            const int kk = key0 + 4 * kc + kb0;
            v2f a;                              // A frag from probabilities in LDS
            a.x = sc[l16 * S_LEN + kk];
            a.y = sc[l16 * S_LEN + kk + 1];
            const float* vp = vh + (long)kk * D_HEAD + l16;
#pragma unroll
            for (int nb = 0; nb < 4; ++nb) {    // N = d chunk of 16
                v2f bf;
                bf.x = vp[nb * 16];
                bf.y = vp[D_HEAD + nb * 16];
                acc[nb] = __builtin_amdgcn_wmma_f32_16x16x4_f32(false, a, false, bf,
                                                                (short)0, acc[nb],
                                                                false, false);
            }
        }
    }
    // cross-wave reduction via LDS float atomics (ds_add_f32)
#pragma unroll
    for (int nb = 0; nb < 4; ++nb) {
#pragma unroll
        for (int r = 0; r < 8; ++r) {
            const int m = r + 8 * half;
            atomicAdd(&outbuf[m * D_HEAD + nb * 16 + l16], acc[nb][r]);
        }
    }
    __syncthreads();

    // ---- Stage 4: output = AV + V (residual) ----
    for (int idx = tid; idx < 16 * D_HEAD; idx += 256) {
        const int m = idx >> 6;
        const int d = idx & 63;
        out[(hbase + row0 + m) * D_HEAD + d] =
            outbuf[idx] + vh[(long)(row0 + m) * D_HEAD + d];
    }
}

extern "C" void kernel_launch(void* const* d_in, const int* in_sizes, int n_in,
                              void* d_out, int out_size, void* d_ws, size_t ws_size,
                              hipStream_t stream) {
    const float* q    = (const float*)d_in[0];
    const float* k    = (const float*)d_in[1];
    const float* v    = (const float*)d_in[2];
    const int*   mask = (const int*)d_in[3];

    float* out  = (float*)d_out;                       // [B,H,S,D]
    float* attn = out + (long)N_BH * S_LEN * D_HEAD;   // [B,H,S,S] follows

    const size_t smem_bytes = (size_t)SMEM_FLOATS * sizeof(float); // ~136 KB (<320 KB WGP LDS)
    hipFuncSetAttribute((const void*)attn_fused_kernel,
                        hipFuncAttributeMaxDynamicSharedMemorySize,
                        (int)smem_bytes);

    dim3 grid(N_BH * (S_LEN / 16));   // 64 heads * 128 row tiles = 8192
    dim3 block(256);                  // 8 waves
    attn_fused_kernel<<<grid, block, smem_bytes, stream>>>(q, k, v, mask, out, attn);
}